// RocAucLoss_532575944928
// MI455X (gfx1250) — compile-verified
//
#include <hip/hip_runtime.h>
#include <hip/hip_bf16.h>

typedef __attribute__((ext_vector_type(2))) float v2f;
typedef __attribute__((ext_vector_type(8))) float v8f;

#define ROC_EPS   1e-4f
#define ROC_GAMMA 0.7f
#define ROC_BIG   1.0e6f   // sentinel: guarantees d > 0 -> min(d,0)^2 == 0 for masked pairs

// ---------------------------------------------------------------------------
// Kernel 1: per-element transform.
//   y = clamp(sigmoid(pred), eps, 1-eps)
//   arrA[i] = (true==0) ? -y : BIG      (this is  -a_i  used directly in A frag)
//   arrB[j] = (true==1) ? y - gamma : BIG
// Pair value becomes  min(arrA[i] + arrB[j], 0)^2  for ALL i,j (mask folded in).
// ---------------------------------------------------------------------------
__global__ void roc_prep_kernel(const float* __restrict__ y_pred,
                                const int* __restrict__ y_true,
                                float* __restrict__ arrA,
                                float* __restrict__ arrB, int n) {
    int i = blockIdx.x * blockDim.x + threadIdx.x;
    if (i >= n) return;
    float y = 1.0f / (1.0f + expf(-y_pred[i]));
    y = fminf(fmaxf(y, ROC_EPS), 1.0f - ROC_EPS);
    int t = y_true[i];
    arrA[i] = (t == 0) ? -y : ROC_BIG;
    arrB[i] = (t == 1) ? (y - ROC_GAMMA) : ROC_BIG;
}

// ---------------------------------------------------------------------------
// Kernel 2: pair sweep. One wave32 owns one 16-row tile; loops over all
// 16-column tiles. Each tile's 16x16 diff matrix comes from ONE
// v_wmma_f32_16x16x4_f32 (rank-2 outer-sum trick), then per lane
// 8x v_med3_f32 (= min(d,0), no NaN-canonicalize) + 4x v_pk_fma_f32.
// A-matrix 16x4 f32 layout: V0 = K0|K2, V1 = K1|K3 (lanes 0-15 | 16-31).
// Row A_i = [-a_i, 1, 0, 0]; Col B_j = [1, b_j - gamma, 0, 0]^T.
// ---------------------------------------------------------------------------
__global__ void roc_pair_kernel(const float* __restrict__ arrA,
                                const float* __restrict__ arrB,
                                float* __restrict__ partials, int n) {
    const int tid    = blockIdx.x * blockDim.x + threadIdx.x;
    const int wave   = tid >> 5;
    const int lane   = tid & 31;
    const int lane16 = lane & 15;
    const bool lo    = (lane < 16);
    const int i0     = wave * 16;
    const float ninf = -__builtin_inff();

    // Loop-invariant A fragment for this wave's 16 rows.
    float aval = arrA[i0 + lane16];       // all lanes load; upper half discards
    v2f afrag;
    afrag.x = lo ? aval : 0.0f;           // K=0 (lanes<16) / K=2 zero
    afrag.y = lo ? 1.0f : 0.0f;           // K=1            / K=3 zero

    const v8f zero = {};
    v8f acc = {};

    const int ntiles = n >> 4;
    for (int jt = 0; jt < ntiles; ++jt) {
        float bv = arrB[(jt << 4) + lane16];
        v2f bfrag;
        bfrag.x = lo ? 1.0f : 0.0f;       // K=0 row of B: all ones
        bfrag.y = lo ? bv   : 0.0f;       // K=1 row of B: b_j - gamma
        // D[i][j] = (-a_i)*1 + 1*(b_j - gamma) = y_j - y_i - gamma
        v8f d = __builtin_amdgcn_wmma_f32_16x16x4_f32(
            /*neg_a=*/false, afrag, /*neg_b=*/false, bfrag,
            /*c_mod=*/(short)0, zero, /*reuse_a=*/false, /*reuse_b=*/false);
#pragma unroll
        for (int v = 0; v < 8; ++v) {
            // med3(d, 0, -inf) == min(d, 0) for ordered d (WMMA output, never sNaN)
            float m = __builtin_amdgcn_fmed3f(d[v], 0.0f, ninf);
            acc[v] = fmaf(m, m, acc[v]);
        }
    }

    float s = 0.0f;
#pragma unroll
    for (int v = 0; v < 8; ++v) s += acc[v];
    // deterministic wave32 butterfly reduce
    for (int off = 16; off > 0; off >>= 1)
        s += __shfl_down(s, off, 32);
    if (lane == 0) partials[wave] = s;
}

// ---------------------------------------------------------------------------
// Kernel 3: deterministic finalize. One block: tree-reduce the wave partials,
// recount P/Ng via sentinel compare (exact, order-free), divide.
// ---------------------------------------------------------------------------
__global__ void roc_finalize_kernel(const float* __restrict__ arrA,
                                    const float* __restrict__ arrB,
                                    const float* __restrict__ partials,
                                    float* __restrict__ out,
                                    int n, int nparts) {
    __shared__ float ssum[256];
    __shared__ int   spos[256];
    __shared__ int   sneg[256];
    const int t = threadIdx.x;

    float s = 0.0f;
    for (int idx = t; idx < nparts; idx += 256) s += partials[idx];
    int p = 0, g = 0;
    for (int idx = t; idx < n; idx += 256) {
        p += (arrB[idx] != ROC_BIG) ? 1 : 0;
        g += (arrA[idx] != ROC_BIG) ? 1 : 0;
    }
    ssum[t] = s; spos[t] = p; sneg[t] = g;
    __syncthreads();
    for (int stride = 128; stride > 0; stride >>= 1) {
        if (t < stride) {
            ssum[t] += ssum[t + stride];
            spos[t] += spos[t + stride];
            sneg[t] += sneg[t + stride];
        }
        __syncthreads();
    }
    if (t == 0) {
        // match jax: (P * Ng) computed exactly, then cast to f32
        float npairs = (float)((double)spos[0] * (double)sneg[0]);
        out[0] = ssum[0] / npairs;
    }
}

extern "C" void kernel_launch(void* const* d_in, const int* in_sizes, int n_in,
                              void* d_out, int out_size, void* d_ws, size_t ws_size,
                              hipStream_t stream) {
    (void)n_in; (void)out_size; (void)ws_size;
    const float* y_pred = (const float*)d_in[0];
    const int*   y_true = (const int*)d_in[1];
    const int n = in_sizes[0];            // 16384

    char* ws = (char*)d_ws;
    float* arrA     = (float*)(ws);
    float* arrB     = (float*)(ws + (size_t)n * sizeof(float));
    float* partials = (float*)(ws + (size_t)2 * n * sizeof(float));

    const int nwaves = n >> 4;            // one wave per 16-row tile (1024)

    roc_prep_kernel<<<(n + 255) / 256, 256, 0, stream>>>(y_pred, y_true, arrA, arrB, n);
    roc_pair_kernel<<<(nwaves * 32) / 256, 256, 0, stream>>>(arrA, arrB, partials, n);
    roc_finalize_kernel<<<1, 256, 0, stream>>>(arrA, arrB, partials, (float*)d_out, n, nwaves);
}